// Pointnet2Partial_6244882448519
// MI455X (gfx1250) — compile-verified
//
#include <hip/hip_runtime.h>
#include <hip/hip_bf16.h>

typedef __attribute__((ext_vector_type(2))) float v2f;
typedef __attribute__((ext_vector_type(8))) float v8f;

#define LDS_FENCE() asm volatile("s_wait_dscnt 0" ::: "memory")

// ---------------------------------------------------------------------------
// Weight prep: W [Nc x K] row-major  ->  Wt [Kp x Nc], zero-padded K rows.
// ---------------------------------------------------------------------------
__global__ void prep_weights_kernel(const float* __restrict__ W, float* __restrict__ Wt,
                                    int Nc, int K, int Kp) {
    int idx = blockIdx.x * blockDim.x + threadIdx.x;
    if (idx >= Kp * Nc) return;
    int k = idx / Nc, n = idx - k * Nc;
    Wt[idx] = (k < K) ? W[n * K + k] : 0.0f;
}

// ---------------------------------------------------------------------------
// Farthest point sampling. One workgroup (1024 threads) per batch.
// Writes selected center coordinates directly (indices never needed later).
// Tie-break matches jnp.argmax (first max) via packed (d2bits, N-1-idx) keys.
// ---------------------------------------------------------------------------
template <int N, int S>
__global__ void fps_kernel(const float* __restrict__ pos, float* __restrict__ centers) {
    constexpr int T = 1024;
    constexpr int PPT = N / T;
    __shared__ float posl[3 * N];
    __shared__ unsigned long long wkeys[T / 32];

    const int tid = threadIdx.x;
    const int b = blockIdx.x;

    for (int i = tid; i < N; i += T) {
        posl[3 * i + 0] = pos[((size_t)b * N + i) * 3 + 0];
        posl[3 * i + 1] = pos[((size_t)b * N + i) * 3 + 1];
        posl[3 * i + 2] = pos[((size_t)b * N + i) * 3 + 2];
    }
    __syncthreads();

    float px[PPT], py[PPT], pz[PPT], d[PPT];
    const float fx = posl[0], fy = posl[1], fz = posl[2];
#pragma unroll
    for (int j = 0; j < PPT; ++j) {
        int i = tid * PPT + j;
        px[j] = posl[3 * i + 0];
        py[j] = posl[3 * i + 1];
        pz[j] = posl[3 * i + 2];
        float dx = px[j] - fx, dy = py[j] - fy, dz = pz[j] - fz;
        d[j] = dx * dx + dy * dy + dz * dz;
    }
    if (tid == 0) {
        centers[(size_t)b * S * 3 + 0] = fx;
        centers[(size_t)b * S * 3 + 1] = fy;
        centers[(size_t)b * S * 3 + 2] = fz;
    }

    for (int s = 1; s < S; ++s) {
        float bv = -1.0f;
        int bi = tid * PPT;
#pragma unroll
        for (int j = 0; j < PPT; ++j)
            if (d[j] > bv) { bv = d[j]; bi = tid * PPT + j; }
        unsigned long long key =
            ((unsigned long long)__float_as_uint(bv) << 32) | (unsigned)(N - 1 - bi);
#pragma unroll
        for (int off = 16; off; off >>= 1) {
            unsigned long long o = __shfl_xor(key, off, 32);
            key = (o > key) ? o : key;
        }
        if ((tid & 31) == 0) wkeys[tid >> 5] = key;
        __syncthreads();
        unsigned long long wk = 0;
#pragma unroll 4
        for (int k2 = 0; k2 < T / 32; ++k2) wk = (wkeys[k2] > wk) ? wkeys[k2] : wk;
        const int nxt = N - 1 - (int)(wk & 0xffffffffu);
        const float nx = posl[3 * nxt + 0], ny = posl[3 * nxt + 1], nz = posl[3 * nxt + 2];
        if (tid == 0) {
            centers[((size_t)b * S + s) * 3 + 0] = nx;
            centers[((size_t)b * S + s) * 3 + 1] = ny;
            centers[((size_t)b * S + s) * 3 + 2] = nz;
        }
#pragma unroll
        for (int j = 0; j < PPT; ++j) {
            float dx = px[j] - nx, dy = py[j] - ny, dz = pz[j] - nz;
            float dd = dx * dx + dy * dy + dz * dz;
            d[j] = fminf(d[j], dd);
        }
        __syncthreads();
    }
}

// ---------------------------------------------------------------------------
// Radius grouping: up to 64 nearest within r per center. One wave per center.
// d2 cached in LDS; >64 in radius -> bisection on distance threshold.
// ---------------------------------------------------------------------------
__global__ void radius_kernel(const float* __restrict__ pos, const float* __restrict__ centers,
                              int* __restrict__ nbr, int* __restrict__ cnt,
                              int N, int M, float r2) {
    extern __shared__ float smem[];
    const int nw = blockDim.x >> 5;
    const int wave = threadIdx.x >> 5;
    const int lane = threadIdx.x & 31;
    float* d2b = smem + (size_t)wave * N;
    int* slot = (int*)(smem + (size_t)nw * N) + wave;

    const int cIdx = blockIdx.x * nw + wave;
    const int b = cIdx / M;
    const float cx = centers[cIdx * 3 + 0];
    const float cy = centers[cIdx * 3 + 1];
    const float cz = centers[cIdx * 3 + 2];
    const float* pb = pos + (size_t)b * N * 3;

    int lc = 0;
    for (int i = lane; i < N; i += 32) {
        float dx = pb[i * 3 + 0] - cx;
        float dy = pb[i * 3 + 1] - cy;
        float dz = pb[i * 3 + 2] - cz;
        float d2 = dx * dx + dy * dy + dz * dz;
        d2b[i] = d2;
        lc += (d2 <= r2) ? 1 : 0;
    }
    LDS_FENCE();
#pragma unroll
    for (int off = 16; off; off >>= 1) lc += __shfl_xor(lc, off, 32);

    float t = r2;
    if (lc > 64) {
        float lo = 0.0f, hi = r2;
        for (int it = 0; it < 36; ++it) {
            float mid = 0.5f * (lo + hi);
            int cc = 0;
            for (int i = lane; i < N; i += 32) cc += (d2b[i] <= mid) ? 1 : 0;
#pragma unroll
            for (int off = 16; off; off >>= 1) cc += __shfl_xor(cc, off, 32);
            if (cc >= 64) hi = mid; else lo = mid;
        }
        t = hi;
    }

    if (lane == 0) *slot = 0;
    LDS_FENCE();
    int* nb = nbr + (size_t)cIdx * 64;
    for (int i = lane; i < N; i += 32) {
        if (d2b[i] <= t) {
            int s = atomicAdd(slot, 1);
            if (s < 64) nb[s] = i;
        }
    }
    LDS_FENCE();
    if (lane == 0) cnt[cIdx] = min(*slot, 64);
}

// ---------------------------------------------------------------------------
// One MLP layer on a 64-row LDS tile using V_WMMA_F32_16X16X4_F32.
// In: LDS [64 x PIN], Wt: global [KP x NL] (K-padded, transposed),
// Out: LDS [64 x NL].  Epilogue: relu(x+b) * (g*rsqrt(1+eps)) + beta.
// ---------------------------------------------------------------------------
template <int KP, int NL, int PIN>
__device__ __forceinline__ void mlp_layer(
    const float* __restrict__ Wt, const float* __restrict__ bias,
    const float* __restrict__ gam, const float* __restrict__ bet,
    const float* In, float* Out, int lane) {
    const int col = lane & 15;
    const int hi2 = (lane >> 4) << 1;  // 0 or 2 (K sub-offset for both halves)
    const float bnr = rsqrtf(1.0f + 1e-5f);
#pragma unroll
    for (int nt = 0; nt < NL / 16; ++nt) {
        v8f acc0 = {}, acc1 = {}, acc2 = {}, acc3 = {};
        const int nc = nt * 16 + col;
        for (int k = 0; k < KP; k += 4) {
            const int kr = k + hi2;
            v2f bf;
            bf.x = Wt[kr * NL + nc];
            bf.y = Wt[(kr + 1) * NL + nc];
            v2f a0 = *(const v2f*)(In + (col)*PIN + kr);
            v2f a1 = *(const v2f*)(In + (16 + col) * PIN + kr);
            v2f a2 = *(const v2f*)(In + (32 + col) * PIN + kr);
            v2f a3 = *(const v2f*)(In + (48 + col) * PIN + kr);
            acc0 = __builtin_amdgcn_wmma_f32_16x16x4_f32(false, a0, false, bf, (short)0, acc0, false, false);
            acc1 = __builtin_amdgcn_wmma_f32_16x16x4_f32(false, a1, false, bf, (short)0, acc1, false, false);
            acc2 = __builtin_amdgcn_wmma_f32_16x16x4_f32(false, a2, false, bf, (short)0, acc2, false, false);
            acc3 = __builtin_amdgcn_wmma_f32_16x16x4_f32(false, a3, false, bf, (short)0, acc3, false, false);
        }
        const float bv = bias[nc];
        const float sc = gam[nc] * bnr;
        const float be = bet[nc];
        const int rbase = (lane >> 4) * 8;
#pragma unroll
        for (int e = 0; e < 8; ++e) {
            int r = rbase + e;
            Out[(r)*NL + nc]      = fmaxf(acc0[e] + bv, 0.0f) * sc + be;
            Out[(16 + r) * NL + nc] = fmaxf(acc1[e] + bv, 0.0f) * sc + be;
            Out[(32 + r) * NL + nc] = fmaxf(acc2[e] + bv, 0.0f) * sc + be;
            Out[(48 + r) * NL + nc] = fmaxf(acc3[e] + bv, 0.0f) * sc + be;
        }
    }
}

// ---------------------------------------------------------------------------
// Fused set-abstraction: gather 64 neighbors -> 3-layer MLP (WMMA) -> maxpool.
// One wave per center; per-wave LDS arena = 64x132 (A) + 64xmax(N1,N3) (B).
// ---------------------------------------------------------------------------
template <int N1, int N2, int N3>
__global__ void sa_fused_kernel(
    const float* __restrict__ x, const float* __restrict__ pos,
    const float* __restrict__ centers, const int* __restrict__ nbr,
    const int* __restrict__ cnt,
    const float* __restrict__ Wt1, const float* __restrict__ b1,
    const float* __restrict__ g1, const float* __restrict__ be1,
    const float* __restrict__ Wt2, const float* __restrict__ b2,
    const float* __restrict__ g2, const float* __restrict__ be2,
    const float* __restrict__ Wt3, const float* __restrict__ b3,
    const float* __restrict__ g3, const float* __restrict__ be3,
    float* __restrict__ out, int M, int Nsrc) {
    constexpr int KP1 = 132;                      // 131 padded to /4
    constexpr int NB = (N1 > N3) ? N1 : N3;
    extern __shared__ float smem[];
    const int wave = threadIdx.x >> 5;
    const int lane = threadIdx.x & 31;
    float* A = smem + (size_t)wave * (64 * KP1 + 64 * NB);
    float* Bb = A + 64 * KP1;

    const int cIdx = blockIdx.x * (blockDim.x >> 5) + wave;
    const int b = cIdx / M;
    const int c = min(cnt[cIdx], 64);
    const float cx = centers[cIdx * 3 + 0];
    const float cy = centers[cIdx * 3 + 1];
    const float cz = centers[cIdx * 3 + 2];
    const int* nb = nbr + (size_t)cIdx * 64;

    // Gather: row r = neighbor r; cols 0..127 = x[nbr], 128..130 = rel pos, 131 = 0.
    for (int r = 0; r < 64; ++r) {
        int j = (r < c) ? nb[r] : -1;
        float4 v = make_float4(0.f, 0.f, 0.f, 0.f);
        if (j >= 0) v = *(const float4*)(x + ((size_t)b * Nsrc + j) * 128 + lane * 4);
        *(float4*)(A + r * KP1 + lane * 4) = v;
        if (lane < 4) {
            float rv = 0.0f;
            if (j >= 0 && lane < 3)
                rv = pos[((size_t)b * Nsrc + j) * 3 + lane] -
                     (lane == 0 ? cx : (lane == 1 ? cy : cz));
            A[r * KP1 + 128 + lane] = rv;
        }
    }
    __syncthreads();
    mlp_layer<KP1, N1, KP1>(Wt1, b1, g1, be1, A, Bb, lane);   // feat -> h1 (B)
    __syncthreads();
    mlp_layer<N1, N2, N1>(Wt2, b2, g2, be2, Bb, A, lane);     // h1 -> h2 (A)
    __syncthreads();
    mlp_layer<N2, N3, N2>(Wt3, b3, g3, be3, A, Bb, lane);     // h2 -> h3 (B)
    __syncthreads();

    // Max-pool over valid neighbor rows; zero if no valid neighbor.
    for (int col = lane; col < N3; col += 32) {
        float r = 0.0f;
        if (c > 0) {
            float mx = -3.4e38f;
            for (int k = 0; k < c; ++k) mx = fmaxf(mx, Bb[k * N3 + col]);
            r = mx;
        }
        out[(size_t)cIdx * N3 + col] = r;
    }
}

__global__ void copy_c1_kernel(const float* __restrict__ src, float* __restrict__ o1,
                               float* __restrict__ o2, int n) {
    int i = blockIdx.x * blockDim.x + threadIdx.x;
    if (i < n) {
        float v = src[i];
        o1[i] = v;
        o2[i] = v;
    }
}

// ---------------------------------------------------------------------------
extern "C" void kernel_launch(void* const* d_in, const int* in_sizes, int n_in,
                              void* d_out, int out_size, void* d_ws, size_t ws_size,
                              hipStream_t stream) {
    (void)in_sizes; (void)n_in; (void)out_size; (void)ws_size;
    constexpr int B = 4, N0 = 4096, M1 = 2048, M2 = 512;

    const float* x0 = (const float*)d_in[0];
    const float* p0 = (const float*)d_in[1];
    const float* x1 = (const float*)d_in[2];
    const float* p1 = (const float*)d_in[3];
    const float *s1W[3], *s1b[3], *s1g[3], *s1be[3];
    const float *s2W[3], *s2b[3], *s2g[3], *s2be[3];
    for (int l = 0; l < 3; ++l) {
        s1W[l] = (const float*)d_in[4 + 4 * l + 0];
        s1b[l] = (const float*)d_in[4 + 4 * l + 1];
        s1g[l] = (const float*)d_in[4 + 4 * l + 2];
        s1be[l] = (const float*)d_in[4 + 4 * l + 3];
        s2W[l] = (const float*)d_in[16 + 4 * l + 0];
        s2b[l] = (const float*)d_in[16 + 4 * l + 1];
        s2g[l] = (const float*)d_in[16 + 4 * l + 2];
        s2be[l] = (const float*)d_in[16 + 4 * l + 3];
    }

    float* ws = (float*)d_ws;
    size_t off = 0;
    auto alloc = [&](size_t nf) { float* p = ws + off; off += nf; return p; };
    float* c0 = alloc((size_t)B * M1 * 3);
    float* c1 = alloc((size_t)B * M2 * 3);
    float* Wt1a = alloc(132 * 64);
    float* Wt2a = alloc(64 * 64);
    float* Wt3a = alloc(64 * 128);
    float* Wt1b = alloc(132 * 128);
    float* Wt2b = alloc(128 * 128);
    float* Wt3b = alloc(128 * 256);
    float* xs10 = alloc((size_t)B * M1 * 128);
    float* xs11 = alloc((size_t)B * M1 * 128);
    int* nbrA = (int*)alloc((size_t)B * M1 * 64);
    int* cntA = (int*)alloc((size_t)B * M1);
    int* nbrB = (int*)alloc((size_t)B * M2 * 64);
    int* cntB = (int*)alloc((size_t)B * M2);
    int* nbrC = (int*)alloc((size_t)B * M1 * 64);
    int* cntC = (int*)alloc((size_t)B * M1);
    int* nbrD = (int*)alloc((size_t)B * M2 * 64);
    int* cntD = (int*)alloc((size_t)B * M2);

    float* outX20 = (float*)d_out;                               // [B,512,256]
    float* outC1a = outX20 + (size_t)B * M2 * 256;               // [B,512,3]
    float* outX21 = outC1a + (size_t)B * M2 * 3;                 // [B,512,128]
    float* outC1b = outX21 + (size_t)B * M2 * 128;               // [B,512,3]

    auto prep = [&](const float* W, float* Wt, int Nc, int K, int Kp) {
        int tot = Kp * Nc;
        prep_weights_kernel<<<(tot + 255) / 256, 256, 0, stream>>>(W, Wt, Nc, K, Kp);
    };
    prep(s1W[0], Wt1a, 64, 131, 132);
    prep(s1W[1], Wt2a, 64, 64, 64);
    prep(s1W[2], Wt3a, 128, 64, 64);
    prep(s2W[0], Wt1b, 128, 131, 132);
    prep(s2W[1], Wt2b, 128, 128, 128);
    prep(s2W[2], Wt3b, 256, 128, 128);

    const size_t sa1Shm = (size_t)4 * (64 * 132 + 64 * 128) * 4;  // 266240
    const size_t sa2Shm = (size_t)2 * (64 * 132 + 64 * 256) * 4;  // 198656
    hipFuncSetAttribute((const void*)sa_fused_kernel<64, 64, 128>,
                        hipFuncAttributeMaxDynamicSharedMemorySize, (int)sa1Shm);
    hipFuncSetAttribute((const void*)sa_fused_kernel<128, 128, 256>,
                        hipFuncAttributeMaxDynamicSharedMemorySize, (int)sa2Shm);
    hipFuncSetAttribute((const void*)radius_kernel,
                        hipFuncAttributeMaxDynamicSharedMemorySize,
                        (int)(4 * (size_t)N0 * 4 + 16));

    // ---- stream 0: sa1 (FPS on pos_0) ----
    fps_kernel<N0, M1><<<B, 1024, 0, stream>>>(p0, c0);
    radius_kernel<<<(B * M1) / 4, 128, 4 * (size_t)N0 * 4 + 16, stream>>>(
        p0, c0, nbrA, cntA, N0, M1, 0.2f * 0.2f);
    sa_fused_kernel<64, 64, 128><<<(B * M1) / 4, 128, sa1Shm, stream>>>(
        x0, p0, c0, nbrA, cntA,
        Wt1a, s1b[0], s1g[0], s1be[0], Wt2a, s1b[1], s1g[1], s1be[1],
        Wt3a, s1b[2], s1g[2], s1be[2], xs10, M1, N0);

    // ---- stream 0: sa2 (FPS on c0) ----
    fps_kernel<M1, M2><<<B, 1024, 0, stream>>>(c0, c1);
    radius_kernel<<<(B * M2) / 4, 128, 4 * (size_t)M1 * 4 + 16, stream>>>(
        c0, c1, nbrB, cntB, M1, M2, 0.4f * 0.4f);
    sa_fused_kernel<128, 128, 256><<<(B * M2) / 2, 64, sa2Shm, stream>>>(
        xs10, c0, c1, nbrB, cntB,
        Wt1b, s2b[0], s2g[0], s2be[0], Wt2b, s2b[1], s2g[1], s2be[1],
        Wt3b, s2b[2], s2g[2], s2be[2], outX20, M2, M1);

    // ---- stream 1: sa1 with centers c0 ----
    radius_kernel<<<(B * M1) / 4, 128, 4 * (size_t)N0 * 4 + 16, stream>>>(
        p1, c0, nbrC, cntC, N0, M1, 0.2f * 0.2f);
    sa_fused_kernel<64, 64, 128><<<(B * M1) / 4, 128, sa1Shm, stream>>>(
        x1, p1, c0, nbrC, cntC,
        Wt1a, s1b[0], s1g[0], s1be[0], Wt2a, s1b[1], s1g[1], s1be[1],
        Wt3a, s1b[2], s1g[2], s1be[2], xs11, M1, N0);

    // ---- stream 1: sa1 again with centers c1 ----
    radius_kernel<<<(B * M2) / 4, 128, 4 * (size_t)M1 * 4 + 16, stream>>>(
        c0, c1, nbrD, cntD, M1, M2, 0.2f * 0.2f);
    sa_fused_kernel<64, 64, 128><<<(B * M2) / 4, 128, sa1Shm, stream>>>(
        xs11, c0, c1, nbrD, cntD,
        Wt1a, s1b[0], s1g[0], s1be[0], Wt2a, s1b[1], s1g[1], s1be[1],
        Wt3a, s1b[2], s1g[2], s1be[2], outX21, M2, M1);

    copy_c1_kernel<<<(B * M2 * 3 + 255) / 256, 256, 0, stream>>>(c1, outC1a, outC1b, B * M2 * 3);
}